// CriticModel_90512140796584
// MI455X (gfx1250) — compile-verified
//
#include <hip/hip_runtime.h>
#include <hip/hip_bf16.h>

typedef __attribute__((ext_vector_type(2))) float v2f;
typedef __attribute__((ext_vector_type(8))) float v8f;
typedef int v4i_t __attribute__((vector_size(16)));

#define AS1 __attribute__((address_space(1)))
#define AS3 __attribute__((address_space(3)))

#define HIDDEN_D 512
#define GOAL_D   128
#define INDIM    640
#define H16      16
#define WAVES_PER_BLOCK 8

// order-preserving float <-> uint key for atomic max (handles negatives)
__device__ __forceinline__ unsigned fkey(float f) {
  int i = __float_as_int(f);
  return (unsigned)(i >= 0 ? (i ^ 0x80000000) : ~i);
}
__device__ __forceinline__ float fdec(unsigned u) {
  int i = (u & 0x80000000u) ? (int)(u ^ 0x80000000u) : ~(int)u;
  return __int_as_float(i);
}

__global__ void init_kernel(float* sums, float* cnts, unsigned* maxu, int nseg) {
  int i = blockIdx.x * blockDim.x + threadIdx.x;
  if (i < nseg) { sums[i] = 0.0f; cnts[i] = 0.0f; maxu[i] = 0u; }
}

__global__ void finalize_kernel(const float* sums, const float* cnts,
                                const unsigned* maxu, float* out, int nseg) {
  int i = blockIdx.x * blockDim.x + threadIdx.x;
  if (i < nseg) {
    float c = cnts[i];
    float mean = sums[i] / fmaxf(c, 1.0f);
    float mx = (c > 0.0f) ? fdec(maxu[i]) : -__builtin_inff();
    out[i] = 0.5f * mx + 0.5f * mean;
  }
}

__global__ __launch_bounds__(256) void critic_mlp_kernel(
    const float* __restrict__ nodes, const float* __restrict__ goal,
    const int* __restrict__ seg, const float* __restrict__ W1,
    const float* __restrict__ b1, const float* __restrict__ W2,
    const float* __restrict__ b2, float* __restrict__ sums,
    float* __restrict__ cnts, unsigned* __restrict__ maxu, int ntiles) {
  // W1 staged raw: [640][16] f32 row-major = 40 KB
  __shared__ float4 ldsW4[INDIM * H16 / 4];
  const float* ldsW = (const float*)ldsW4;

  {
    const float4* g4 = (const float4*)W1;
#if __has_builtin(__builtin_amdgcn_global_load_async_to_lds_b128)
    for (int j = threadIdx.x; j < INDIM * H16 / 4; j += 256) {
      __builtin_amdgcn_global_load_async_to_lds_b128(
          (AS1 v4i_t*)(g4 + j),
          (AS3 v4i_t*)(ldsW4 + j), 0, 0);
    }
    asm volatile("s_wait_asynccnt 0" ::: "memory");
#else
    for (int j = threadIdx.x; j < INDIM * H16 / 4; j += 256) ldsW4[j] = g4[j];
#endif
  }
  __syncthreads();

  const int wave = threadIdx.x >> 5;
  const int lane = threadIdx.x & 31;
  const int tile = blockIdx.x * WAVES_PER_BLOCK + wave;
  if (tile >= ntiles) return;           // whole-wave uniform: EXEC all-ones inside

  const int col  = lane & 15;           // WMMA column / node-row within tile
  const int half = lane >> 4;           // K sub-pair select per ISA 16x4 A layout
  const int row  = tile * 16 + col;

  const float* nrow = nodes + (long long)row * HIDDEN_D;
  const float* grow = goal  + (long long)row * GOAL_D;

  v8f c = {};
  // ---- nodes part of concat: K = 0..511 ----
#pragma unroll 8
  for (int k = 0; k < HIDDEN_D; k += 4) {
    const int ka = k + 2 * half;
    float2 av = *(const float2*)(nrow + ka);
    v2f a; a.x = av.x; a.y = av.y;
    v2f bb; bb.x = ldsW[ka * H16 + col]; bb.y = ldsW[(ka + 1) * H16 + col];
    c = __builtin_amdgcn_wmma_f32_16x16x4_f32(false, a, false, bb, (short)0, c,
                                              false, false);
  }
  // ---- goal part of concat: K = 512..639 ----
#pragma unroll 8
  for (int k = 0; k < GOAL_D; k += 4) {
    const int ka = k + 2 * half;
    float2 av = *(const float2*)(grow + ka);
    v2f a; a.x = av.x; a.y = av.y;
    v2f bb; bb.x = ldsW[(HIDDEN_D + ka) * H16 + col];
    bb.y = ldsW[(HIDDEN_D + ka + 1) * H16 + col];
    c = __builtin_amdgcn_wmma_f32_16x16x4_f32(false, a, false, bb, (short)0, c,
                                              false, false);
  }

  // ---- epilogue: bias, relu, dot with W2, half-wave butterfly reduce ----
  const float b1v = b1[col];
  const float w2v = W2[col];
  const float b2v = b2[0];

  float o[8];
#pragma unroll
  for (int r = 0; r < 8; ++r) {
    float t = c[r] + b1v;
    t = fmaxf(t, 0.0f) * w2v;
    t += __shfl_xor(t, 1, 32);
    t += __shfl_xor(t, 2, 32);
    t += __shfl_xor(t, 4, 32);
    t += __shfl_xor(t, 8, 32);          // masks <16: stays within half-wave
    o[r] = t + b2v;                     // lanes<16: out[tile*16+r]; lanes>=16: +8
  }

  float lsum = 0.0f, lmax = -__builtin_inff();
#pragma unroll
  for (int r = 0; r < 8; ++r) { lsum += o[r]; lmax = fmaxf(lmax, o[r]); }
  const float osum = lsum + __shfl_xor(lsum, 16, 32);
  const float omax = fmaxf(lmax, __shfl_xor(lmax, 16, 32));

  // segment ids are sorted -> most tiles are single-segment
  const int sid = seg[tile * 16 + col];
  int mn = sid, mx = sid;
#pragma unroll
  for (int m = 1; m <= 8; m <<= 1) {
    mn = min(mn, __shfl_xor(mn, m, 32));
    mx = max(mx, __shfl_xor(mx, m, 32));
  }

  if (mn == mx) {
    if (lane == 0) {
      atomicAdd(&sums[sid], osum);
      atomicAdd(&cnts[sid], 16.0f);
      atomicMax(&maxu[sid], fkey(omax));
    }
  } else {
    float mv = 0.0f;
#pragma unroll
    for (int r = 0; r < 8; ++r)
      if ((lane & 7) == r) mv = o[r];
    if (!(lane & 8)) {                  // 16 distinct nodes: lanes 0-7 & 16-23
      const int node = tile * 16 + half * 8 + (lane & 7);
      const int id2 = seg[node];
      atomicAdd(&sums[id2], mv);
      atomicAdd(&cnts[id2], 1.0f);
      atomicMax(&maxu[id2], fkey(mv));
    }
  }
}

extern "C" void kernel_launch(void* const* d_in, const int* in_sizes, int n_in,
                              void* d_out, int out_size, void* d_ws, size_t ws_size,
                              hipStream_t stream) {
  const float* nodes = (const float*)d_in[0];
  const float* goal  = (const float*)d_in[1];
  const int*   seg   = (const int*)d_in[2];
  // d_in[3] = num_segments scalar (device side); use out_size instead
  const float* W1 = (const float*)d_in[4];
  const float* b1 = (const float*)d_in[5];
  const float* W2 = (const float*)d_in[6];
  const float* b2 = (const float*)d_in[7];

  const int nseg = out_size;
  const int N = in_sizes[0] / HIDDEN_D;
  const int ntiles = (N + 15) / 16;

  float* sums = (float*)d_ws;
  float* cnts = sums + nseg;
  unsigned* maxu = (unsigned*)(cnts + nseg);

  init_kernel<<<(nseg + 255) / 256, 256, 0, stream>>>(sums, cnts, maxu, nseg);
  critic_mlp_kernel<<<(ntiles + WAVES_PER_BLOCK - 1) / WAVES_PER_BLOCK, 256, 0,
                      stream>>>(nodes, goal, seg, W1, b1, W2, b2, sums, cnts,
                                maxu, ntiles);
  finalize_kernel<<<(nseg + 255) / 256, 256, 0, stream>>>(sums, cnts, maxu,
                                                          (float*)d_out, nseg);
}